// MPODecomposition4D_8581344657408
// MI455X (gfx1250) — compile-verified
//
#include <hip/hip_runtime.h>
#include <hip/hip_bf16.h>
#include <stdint.h>

// Problem constants (from reference)
#define TT 1000
#define XX 512
#define YY 512
#define UU 2
#define RT 8
#define RS 12
#define RP 4

typedef __attribute__((ext_vector_type(2))) float v2f;
typedef __attribute__((ext_vector_type(8))) float v8f;

// ---------------------------------------------------------------------------
// Pass 1: W[t][u][j] = sum_{i,k} edge_time[t,i] * core[i,j,k] * ep[u,k]
// Formulated as GEMM per u:  W_u (1000x12) = ET (1000x8) @ CP_u (8x12)
// One wave computes a 16x16 output tile via two V_WMMA_F32_16X16X4_F32
// (K = 8 split into 4 + 4).
// ---------------------------------------------------------------------------
__global__ __launch_bounds__(32) void mpo_pass1_wmma(
    const float* __restrict__ ET,    // (1000, 8)
    const float* __restrict__ core,  // (8, 12, 4)
    const float* __restrict__ EP,    // (2, 4)
    float* __restrict__ W)           // (1000, 2, 12)
{
    const int m0   = blockIdx.x * 16;      // tile row base (0..62)
    const int u    = blockIdx.y;           // 0..1
    const int lane = threadIdx.x;          // 0..31
    const int half = lane >> 4;            // 0 or 1
    const int l    = lane & 15;

    // ---- A matrix (16x4 f32): lane L<16 holds rows M=L, K={0,1} in v0,v1;
    //      lanes 16-31 hold K={2,3}. Second WMMA covers K=4..7.
    int row = m0 + l;
    row = (row < TT) ? row : (TT - 1);     // clamp (OOB rows masked at store)
    v2f a0, a1;
    a0.x = ET[row * RT + 2 * half + 0];
    a0.y = ET[row * RT + 2 * half + 1];
    a1.x = ET[row * RT + 4 + 2 * half + 0];
    a1.y = ET[row * RT + 4 + 2 * half + 1];

    // ---- B matrix (4x16 f32): CP_u[i][j] = core[i,j,:] . ep[u,:]
    //      VGPR0 (b.x): rows K = 2*half ; VGPR1 (b.y): rows K = 2*half+1
    const float ep0 = EP[u * RP + 0], ep1 = EP[u * RP + 1];
    const float ep2 = EP[u * RP + 2], ep3 = EP[u * RP + 3];
    auto cp = [&](int i, int j) -> float {
        if (j >= RS) return 0.0f;          // pad N cols 12..15 with zeros
        const float* c = core + (i * RS + j) * RP;
        return c[0] * ep0 + c[1] * ep1 + c[2] * ep2 + c[3] * ep3;
    };
    v2f b0, b1;
    b0.x = cp(2 * half + 0, l);
    b0.y = cp(2 * half + 1, l);
    b1.x = cp(4 + 2 * half + 0, l);
    b1.y = cp(4 + 2 * half + 1, l);

    v8f c = {};
    c = __builtin_amdgcn_wmma_f32_16x16x4_f32(false, a0, false, b0,
                                              (short)0, c, false, false);
    c = __builtin_amdgcn_wmma_f32_16x16x4_f32(false, a1, false, b1,
                                              (short)0, c, false, false);

    // ---- Store C tile: VGPR r, lane: row M = r + 8*half, col N = l
    if (l < RS) {
        #pragma unroll
        for (int r = 0; r < 8; ++r) {
            int m = m0 + r + 8 * half;
            if (m < TT) W[(m * UU + u) * RS + l] = c[r];
        }
    }
}

// ---------------------------------------------------------------------------
// Pass 2: out[b] = sum_j W[t,u,j] * SX[x,j] * SY[y,j]
// Memory-bound: 16B idx read + 4B write per sample (~80 MB total @ 23.3 TB/s).
// W (96 KB) + SX (24 KB) + SY (24 KB) staged into LDS with CDNA5 async
// global->LDS b128 copies (ASYNCcnt), then pure LDS gather + VALU.
// ---------------------------------------------------------------------------
#define LDS_W_OFF  0u
#define LDS_SX_OFF 96000u                 // 1000*2*12*4
#define LDS_SY_OFF 120576u                // + 512*12*4
#define LDS_BYTES  145152u                // + 512*12*4
#define ITEMS 16

__device__ __forceinline__ void stage_async_b128(const float* __restrict__ src,
                                                 uint32_t lds_off,
                                                 uint32_t nbytes,
                                                 uint32_t tid,
                                                 uint32_t nthreads)
{
    const uint32_t chunks = nbytes >> 4;
    for (uint32_t ch = tid; ch < chunks; ch += nthreads) {
        uint32_t laddr = lds_off + (ch << 4);   // LDS byte address (VDST vgpr)
        uint32_t goff  = ch << 4;               // 32-bit vaddr offset (GVS)
        asm volatile("global_load_async_to_lds_b128 %0, %1, %2"
                     :: "v"(laddr), "v"(goff), "s"(src)
                     : "memory");
    }
}

__global__ __launch_bounds__(256) void mpo_pass2_gather(
    const int*   __restrict__ idx,   // (B, 4): t, x, y, u
    const float* __restrict__ W,     // (1000, 2, 12) from pass 1 (d_ws)
    const float* __restrict__ SX,    // (512, 12)
    const float* __restrict__ SY,    // (512, 12)
    float*       __restrict__ out,   // (B,)
    int Btot)
{
    extern __shared__ char smem[];
    const float* sW  = (const float*)(smem + LDS_W_OFF);
    const float* sSX = (const float*)(smem + LDS_SX_OFF);
    const float* sSY = (const float*)(smem + LDS_SY_OFF);

    const uint32_t tid = threadIdx.x;

    // ---- async-stage all tables into LDS (ASYNCcnt-tracked) ----
    stage_async_b128(W,  LDS_W_OFF,  TT * UU * RS * 4, tid, 256);
    stage_async_b128(SX, LDS_SX_OFF, XX * RS * 4,      tid, 256);
    stage_async_b128(SY, LDS_SY_OFF, YY * RS * 4,      tid, 256);
    asm volatile("s_wait_asynccnt 0" ::: "memory");
    __syncthreads();

    // ---- stream samples: coalesced int4 loads, LDS gathers, 1 f32 store ----
    const int base = blockIdx.x * (256 * ITEMS);
    #pragma unroll 4
    for (int k = 0; k < ITEMS; ++k) {
        const int b = base + k * 256 + (int)tid;
        if (b >= Btot) break;
        const int4 iv = ((const int4*)idx)[b];  // t, x, y, u

        const float4* wr = (const float4*)(sW  + (iv.x * UU + iv.w) * RS);
        const float4* xr = (const float4*)(sSX + iv.y * RS);
        const float4* yr = (const float4*)(sSY + iv.z * RS);

        float4 w0 = wr[0], w1 = wr[1], w2 = wr[2];
        float4 x0 = xr[0], x1 = xr[1], x2 = xr[2];
        float4 y0 = yr[0], y1 = yr[1], y2 = yr[2];

        float acc;
        acc  =      w0.x * (x0.x * y0.x);
        acc = fmaf(w0.y, x0.y * y0.y, acc);
        acc = fmaf(w0.z, x0.z * y0.z, acc);
        acc = fmaf(w0.w, x0.w * y0.w, acc);
        acc = fmaf(w1.x, x1.x * y1.x, acc);
        acc = fmaf(w1.y, x1.y * y1.y, acc);
        acc = fmaf(w1.z, x1.z * y1.z, acc);
        acc = fmaf(w1.w, x1.w * y1.w, acc);
        acc = fmaf(w2.x, x2.x * y2.x, acc);
        acc = fmaf(w2.y, x2.y * y2.y, acc);
        acc = fmaf(w2.z, x2.z * y2.z, acc);
        acc = fmaf(w2.w, x2.w * y2.w, acc);

        out[b] = acc;
    }
}

// ---------------------------------------------------------------------------
extern "C" void kernel_launch(void* const* d_in, const int* in_sizes, int n_in,
                              void* d_out, int out_size, void* d_ws, size_t ws_size,
                              hipStream_t stream) {
    const int*   indices = (const int*)  d_in[0];  // (B,4)
    const float* core    = (const float*)d_in[1];  // (8,12,4)
    const float* ET      = (const float*)d_in[2];  // (1000,8)
    const float* SX      = (const float*)d_in[3];  // (512,12)
    const float* SY      = (const float*)d_in[4];  // (512,12)
    const float* EP      = (const float*)d_in[5];  // (2,4)
    float* out = (float*)d_out;
    float* W   = (float*)d_ws;                     // 96 KB scratch table

    const int Btot = in_sizes[0] / 4;

    // Pass 1: 63 M-tiles x 2 physics channels, one wave each (WMMA GEMM)
    dim3 g1((TT + 15) / 16, UU);
    mpo_pass1_wmma<<<g1, 32, 0, stream>>>(ET, core, EP, W);

    // Pass 2: memory-bound gather stream
    const int itemsPerBlock = 256 * ITEMS;
    const int nb = (Btot + itemsPerBlock - 1) / itemsPerBlock;
    mpo_pass2_gather<<<nb, 256, LDS_BYTES, stream>>>(indices, W, SX, SY, out, Btot);
}